// GraphNet2_10316511445510
// MI455X (gfx1250) — compile-verified
//
#include <hip/hip_runtime.h>
#include <hip/hip_bf16.h>
#include <math.h>

typedef __attribute__((ext_vector_type(2))) float v2f;
typedef __attribute__((ext_vector_type(8))) float v8f;

#define FDIM 128
#define DENC 32

// ---------------------------------------------------------------------------
// Kernel 1: vodes = relu(nodes @ W_enc + b_enc)  via V_WMMA_F32_16X16X4_F32
// One wave (32 lanes) computes a 16x32 output tile (two 16x16 WMMA tiles),
// looping K=0..127 in steps of 4.
// A 16x4 f32 layout:  lanes 0-15 hold M=0..15; vgpr0=K(+2*half), vgpr1=K+1(+2*half)
// B 4x16 f32 layout:  lanes hold N; vgpr0=K(+2*half), vgpr1=K+1(+2*half)
// C/D 16x16 f32:      vgpr r: lanes0-15 -> M=r, lanes16-31 -> M=r+8, N=lane%16
// ---------------------------------------------------------------------------
__global__ void encode_wmma_kernel(const float* __restrict__ nodes,
                                   const float* __restrict__ W_enc,   // [128][32]
                                   const float* __restrict__ b_enc,   // [32]
                                   float* __restrict__ vodes,         // [n][32]
                                   int n_nodes, int num_tiles)
{
    const int wave = threadIdx.x >> 5;
    const int tile = blockIdx.x * (blockDim.x >> 5) + wave;
    if (tile >= num_tiles) return;            // wave-uniform: survivors keep EXEC all-ones
    const int lane = threadIdx.x & 31;
    const int half = lane >> 4;               // 0: lanes 0-15, 1: lanes 16-31
    const int l    = lane & 15;
    const int row0 = tile * 16;

    // clamp the load row so EXEC can stay full even on a ragged tail
    int arow = row0 + l;
    if (arow > n_nodes - 1) arow = n_nodes - 1;
    const float* __restrict__ aBase = nodes + (size_t)arow * FDIM;

    v8f acc0 = {0.f, 0.f, 0.f, 0.f, 0.f, 0.f, 0.f, 0.f};
    v8f acc1 = {0.f, 0.f, 0.f, 0.f, 0.f, 0.f, 0.f, 0.f};

    for (int k = 0; k < FDIM; k += 4) {
        v2f a;
        a.x = aBase[k + 2 * half + 0];
        a.y = aBase[k + 2 * half + 1];

        const float* bp = W_enc + (size_t)(k + 2 * half) * DENC + l;
        v2f b0; b0.x = bp[0];  b0.y = bp[DENC];
        v2f b1; b1.x = bp[16]; b1.y = bp[DENC + 16];

        acc0 = __builtin_amdgcn_wmma_f32_16x16x4_f32(false, a, false, b0,
                                                     (short)0, acc0, false, false);
        acc1 = __builtin_amdgcn_wmma_f32_16x16x4_f32(false, a, false, b1,
                                                     (short)0, acc1, false, false);
    }

    const float bc0 = b_enc[l];
    const float bc1 = b_enc[l + 16];
#pragma unroll
    for (int r = 0; r < 8; ++r) {
        int row = row0 + r + 8 * half;
        if (row < n_nodes) {
            float* out = vodes + (size_t)row * DENC;
            out[l]      = fmaxf(acc0[r] + bc0, 0.f);
            out[l + 16] = fmaxf(acc1[r] + bc1, 0.f);
        }
    }
}

// ---------------------------------------------------------------------------
// Kernel 2: acc = vodes (scatter-add base), zero the pooled[128] accumulator
// ---------------------------------------------------------------------------
__global__ void init_acc_kernel(const float* __restrict__ vodes,
                                float* __restrict__ acc,
                                float* __restrict__ pooled, int total)
{
    int i = blockIdx.x * blockDim.x + threadIdx.x;
    if (i < total) acc[i] = vodes[i];
    if (i < FDIM)  pooled[i] = 0.f;
}

// ---------------------------------------------------------------------------
// Kernel 3: segment_sum — one thread per (edge, channel).
// Gather of vodes[s] is coalesced (a wave covers all 32 channels of one edge);
// atomics land in the L2-resident acc buffer (6.4 MB << 192 MB L2).
// ---------------------------------------------------------------------------
__global__ void scatter_kernel(const float* __restrict__ vodes,
                               const int* __restrict__ senders,
                               const int* __restrict__ receivers,
                               float* __restrict__ acc, long long total)
{
    long long t = (long long)blockIdx.x * blockDim.x + threadIdx.x;
    if (t >= total) return;
    int e = (int)(t >> 5);
    int c = (int)(t & 31);
    int s = senders[e];
    int r = receivers[e];
    atomicAdd(&acc[(size_t)r * DENC + c], vodes[(size_t)s * DENC + c]);
}

// ---------------------------------------------------------------------------
// Kernel 4: influence[i] = [acc_i | nodes_i[127]] . W_inf + b_inf
// ---------------------------------------------------------------------------
__global__ void influence_kernel(const float* __restrict__ acc,
                                 const float* __restrict__ nodes,
                                 const float* __restrict__ W_inf,  // [33]
                                 const float* __restrict__ b_inf,  // [1]
                                 float* __restrict__ infl, int n)
{
    int i = blockIdx.x * blockDim.x + threadIdx.x;
    if (i >= n) return;
    const float* v = acc + (size_t)i * DENC;
    float s = b_inf[0] + nodes[(size_t)i * FDIM + (FDIM - 1)] * W_inf[DENC];
#pragma unroll
    for (int k = 0; k < DENC; ++k) s += v[k] * W_inf[k];
    infl[i] = s;
}

// ---------------------------------------------------------------------------
// Kernel 5: single-block softmax statistics over n scores: scal = {max, sum exp}
// ---------------------------------------------------------------------------
__global__ void softmax_stats_kernel(const float* __restrict__ infl,
                                     float* __restrict__ scal, int n)
{
    __shared__ float red[1024];
    const int t = threadIdx.x;
    float m = -3.402823466e38f;
    for (int i = t; i < n; i += blockDim.x) m = fmaxf(m, infl[i]);
    red[t] = m; __syncthreads();
    for (int s = blockDim.x >> 1; s > 0; s >>= 1) {
        if (t < s) red[t] = fmaxf(red[t], red[t + s]);
        __syncthreads();
    }
    const float mx = red[0];
    __syncthreads();
    float a = 0.f;
    for (int i = t; i < n; i += blockDim.x) a += expf(infl[i] - mx);
    red[t] = a; __syncthreads();
    for (int s = blockDim.x >> 1; s > 0; s >>= 1) {
        if (t < s) red[t] += red[t + s];
        __syncthreads();
    }
    if (t == 0) { scal[0] = mx; scal[1] = red[0]; }
}

// ---------------------------------------------------------------------------
// Kernel 6: pooled[f] = sum_i softmax_i * nodes[i][f]
// (uses  sum_i a_i (x_i W_emb + b) = (sum_i a_i x_i) W_emb + b  since sum a_i = 1)
// ---------------------------------------------------------------------------
__global__ void pool_kernel(const float* __restrict__ nodes,
                            const float* __restrict__ infl,
                            const float* __restrict__ scal,
                            float* __restrict__ pooled, int n)
{
    const int t = threadIdx.x;           // feature 0..127
    const float mx   = scal[0];
    const float rden = 1.0f / scal[1];
    float p = 0.f;
    for (int i = blockIdx.x; i < n; i += gridDim.x) {
        float w = expf(infl[i] - mx) * rden;
        p += w * nodes[(size_t)i * FDIM + t];
    }
    atomicAdd(&pooled[t], p);
}

// ---------------------------------------------------------------------------
// Kernel 7: tiny MLP head on a single block. out = [logits/10 (4), value (1)]
// ---------------------------------------------------------------------------
__global__ void head_kernel(const float* __restrict__ pooled,
                            const float* __restrict__ W_emb, const float* __restrict__ b_emb,
                            const float* __restrict__ W1, const float* __restrict__ b1,
                            const float* __restrict__ W2, const float* __restrict__ b2,
                            const float* __restrict__ Wy, const float* __restrict__ by,
                            const float* __restrict__ Wx, const float* __restrict__ bx,
                            float* __restrict__ out)
{
    __shared__ float g[12];
    __shared__ float h1[128];
    __shared__ float h2[128];
    const int t = threadIdx.x;           // 128 threads
    if (t < 12) {
        float s = b_emb[t];
        for (int k = 0; k < FDIM; ++k) s += pooled[k] * W_emb[k * 12 + t];
        g[t] = s;
    }
    __syncthreads();
    {
        float s = b1[t];
#pragma unroll
        for (int k = 0; k < 12; ++k) s += g[k] * W1[k * 128 + t];
        h1[t] = fmaxf(s, 0.f);
    }
    __syncthreads();
    {
        float s = b2[t];
        for (int k = 0; k < 128; ++k) s += h1[k] * W2[k * 128 + t];
        h2[t] = fmaxf(s, 0.f);
    }
    __syncthreads();
    if (t < 4) {
        float s = bx[t];
        for (int k = 0; k < 128; ++k) s += h2[k] * Wx[k * 4 + t];
        out[t] = s / 10.0f;              // TEMPERATURE
    }
    if (t == 4) {
        float s = by[0];
        for (int k = 0; k < 128; ++k) s += h2[k] * Wy[k];
        out[4] = s;
    }
}

// ---------------------------------------------------------------------------
extern "C" void kernel_launch(void* const* d_in, const int* in_sizes, int n_in,
                              void* d_out, int out_size, void* d_ws, size_t ws_size,
                              hipStream_t stream)
{
    const float* nodes     = (const float*)d_in[0];
    const int*   senders   = (const int*)  d_in[1];
    const int*   receivers = (const int*)  d_in[2];
    const float* W_enc     = (const float*)d_in[3];
    const float* b_enc     = (const float*)d_in[4];
    const float* W_inf     = (const float*)d_in[5];
    const float* b_inf     = (const float*)d_in[6];
    const float* W_emb     = (const float*)d_in[7];
    const float* b_emb     = (const float*)d_in[8];
    const float* W1        = (const float*)d_in[9];
    const float* b1        = (const float*)d_in[10];
    const float* W2        = (const float*)d_in[11];
    const float* b2        = (const float*)d_in[12];
    const float* Wy        = (const float*)d_in[13];
    const float* by        = (const float*)d_in[14];
    const float* Wx        = (const float*)d_in[15];
    const float* bx        = (const float*)d_in[16];
    float* out = (float*)d_out;

    const int n = in_sizes[0] / FDIM;     // 50000 nodes
    const int e = in_sizes[1];            // 1600000 edges

    // workspace layout (floats): vodes[n*32] | acc[n*32] | infl[n] | pooled[128] | scal[2]
    float* ws     = (float*)d_ws;
    float* vodes  = ws;
    float* acc    = vodes + (size_t)n * DENC;
    float* infl   = acc   + (size_t)n * DENC;
    float* pooled = infl  + n;
    float* scal   = pooled + FDIM;

    // 1) encode (WMMA f32)
    const int numTiles = (n + 15) / 16;
    const int wavesPerBlk = 8;            // 256 threads = 8 wave32
    dim3 encBlk(32 * wavesPerBlk);
    dim3 encGrd((numTiles + wavesPerBlk - 1) / wavesPerBlk);
    encode_wmma_kernel<<<encGrd, encBlk, 0, stream>>>(nodes, W_enc, b_enc, vodes, n, numTiles);

    // 2) acc = vodes ; pooled = 0
    {
        int total = n * DENC;
        init_acc_kernel<<<(total + 255) / 256, 256, 0, stream>>>(vodes, acc, pooled, total);
    }

    // 3) edge scatter-add
    {
        long long total = (long long)e * DENC;
        unsigned blocks = (unsigned)((total + 255) / 256);
        scatter_kernel<<<blocks, 256, 0, stream>>>(vodes, senders, receivers, acc, total);
    }

    // 4) influence scores
    influence_kernel<<<(n + 255) / 256, 256, 0, stream>>>(acc, nodes, W_inf, b_inf, infl, n);

    // 5) softmax stats (max, sum-exp)
    softmax_stats_kernel<<<1, 1024, 0, stream>>>(infl, scal, n);

    // 6) attention-weighted feature pooling
    pool_kernel<<<512, 128, 0, stream>>>(nodes, infl, scal, pooled, n);

    // 7) MLP head -> out[5]
    head_kernel<<<1, 128, 0, stream>>>(pooled, W_emb, b_emb, W1, b1, W2, b2,
                                       Wy, by, Wx, bx, out);
}